// NTMMemory_43499428774177
// MI455X (gfx1250) — compile-verified
//
#include <hip/hip_runtime.h>
#include <hip/hip_bf16.h>
#include <math.h>

typedef float v2f __attribute__((ext_vector_type(2)));
typedef float v8f __attribute__((ext_vector_type(8)));

#define BATCH   128
#define RROWS   8
#define NSLOT   4096
#define MDIM    64
#define NTILES  (NSLOT / 16)   // 256
#define COS_EPS 1e-8f
#define ADD_EPS 1e-16f

// One workgroup (256 threads = 8 waves) per batch b.
// Phase 1: WMMA f32 16x16x4 computes dot(keys, memory rows) for all 8 heads,
//          fused with per-row norm; scores beta*cos stored in LDS (8x4096 f32 = 128 KB).
// Phase 2: wave r does softmax / gate / circular shift / sharpen / normalize for head r.
__global__ __launch_bounds__(256)
void ntm_address_kernel(const float* __restrict__ memory,
                        const float* __restrict__ key,
                        const float* __restrict__ beta,
                        const float* __restrict__ gate,
                        const float* __restrict__ shift,
                        const float* __restrict__ gamma,
                        const float* __restrict__ w_prev,
                        float* __restrict__ w_out)
{
    __shared__ float smem[RROWS * NSLOT];   // 128 KB score/work buffer

    const int b    = blockIdx.x;
    const int tid  = threadIdx.x;
    const int lane = tid & 31;
    const int wave = tid >> 5;
    const int half = lane >> 4;   // 0 -> K pair {0,1}, 1 -> K pair {2,3} within each K=4 step
    const int mrow = lane & 15;   // A: head row (0..15, 8..15 zero-padded); B: memory row in tile

    // ---------------- Phase 0: keys (A matrix), key norms, beta ----------------
    const float* kb = key + (size_t)b * RROWS * MDIM;

    v2f a[16];
#pragma unroll
    for (int s = 0; s < 16; ++s) {
        if (mrow < RROWS) {
            const int m = 4 * s + 2 * half;
            a[s].x = kb[mrow * MDIM + m]     + ADD_EPS;
            a[s].y = kb[mrow * MDIM + m + 1] + ADD_EPS;
        } else {
            a[s].x = 0.0f;
            a[s].y = 0.0f;
        }
    }

    float knl = 0.0f;
    if (lane < RROWS) {
        for (int m = 0; m < MDIM; ++m) {
            const float v = kb[lane * MDIM + m] + ADD_EPS;
            knl += v * v;
        }
        knl = sqrtf(knl);
    }
    float kn[RROWS], bet[RROWS];
#pragma unroll
    for (int j = 0; j < RROWS; ++j) {
        kn[j]  = __shfl(knl, j, 32);
        bet[j] = beta[b * RROWS + j];
    }

    // ---------------- Phase 1: WMMA dot + row norms -> LDS scores ----------------
    const float* mb = memory + (size_t)b * NSLOT * MDIM;

    for (int tile = wave; tile < NTILES; tile += 8) {
        const int    n0   = tile * 16;
        const float* rowp = mb + (size_t)(n0 + mrow) * MDIM + 2 * half;

        // prefetch this wave's next tile (8 tiles ahead)
        __builtin_prefetch(rowp + 8 * 16 * MDIM, 0, 1);

        v8f   acc = {};
        float ssq = 0.0f;
#pragma unroll
        for (int s = 0; s < 16; ++s) {
            v2f bv;
            bv.x = rowp[4 * s]     + ADD_EPS;
            bv.y = rowp[4 * s + 1] + ADD_EPS;
            ssq += bv.x * bv.x + bv.y * bv.y;
            // D = A(16x4) * B(4x16) + C  -> v_wmma_f32_16x16x4_f32
            acc = __builtin_amdgcn_wmma_f32_16x16x4_f32(
                      false, a[s], false, bv, (short)0, acc, false, false);
        }

        // lanes L and L^16 jointly hold one full memory row
        ssq += __shfl_xor(ssq, 16, 32);
        const float mem_norm = sqrtf(ssq);

        if (lane < 16) {
            const int n = n0 + lane;
#pragma unroll
            for (int j = 0; j < RROWS; ++j) {
                const float denom = fmaxf(mem_norm * kn[j], COS_EPS);
                smem[j * NSLOT + n] = bet[j] * (acc[j] / denom);
            }
        }
    }

    __syncthreads();

    // ---------------- Phase 2: per-head softmax / gate / shift / sharpen ----------------
    {
        const int r = wave;                 // 8 waves == 8 heads
        float* trow = smem + r * NSLOT;

        // softmax max
        float mx = -INFINITY;
        for (int n = lane; n < NSLOT; n += 32) mx = fmaxf(mx, trow[n]);
#pragma unroll
        for (int o = 16; o > 0; o >>= 1) mx = fmaxf(mx, __shfl_xor(mx, o, 32));

        // softmax sum (store exp in place)
        float Z = 0.0f;
        for (int n = lane; n < NSLOT; n += 32) {
            const float e = __expf(trow[n] - mx);
            trow[n] = e;
            Z += e;
        }
#pragma unroll
        for (int o = 16; o > 0; o >>= 1) Z += __shfl_xor(Z, o, 32);
        const float invZ = 1.0f / Z;

        const int    hr = b * RROWS + r;
        const float  gr = gate[hr];
        const float  s0 = shift[hr * 3 + 0];
        const float  s1 = shift[hr * 3 + 1];
        const float  s2 = shift[hr * 3 + 2];
        const float  gm = gamma[hr];
        const float* wp = w_prev + (size_t)hr * NSLOT;

        // gated interpolation: wg stored back into LDS row
        for (int n = lane; n < NSLOT; n += 32) {
            const float wc = trow[n] * invZ;
            trow[n] = gr * wc + (1.0f - gr) * wp[n];
        }

        __syncthreads();   // make wg visible before neighbor reads (uniform across waves)

        // circular shift + sharpen: pass 1 accumulates sum of c^gamma
        float S = 0.0f;
        for (int n = lane; n < NSLOT; n += 32) {
            const int nm = (n + NSLOT - 1) & (NSLOT - 1);
            const int np = (n + 1) & (NSLOT - 1);
            const float c = s0 * trow[nm] + s1 * trow[n] + s2 * trow[np];
            S += __powf(c, gm);
        }
#pragma unroll
        for (int o = 16; o > 0; o >>= 1) S += __shfl_xor(S, o, 32);
        const float invS = 1.0f / (S + ADD_EPS);

        // pass 2: recompute (avoids LDS WAR hazard) and write normalized weights
        float* wo = w_out + (size_t)hr * NSLOT;
        for (int n = lane; n < NSLOT; n += 32) {
            const int nm = (n + NSLOT - 1) & (NSLOT - 1);
            const int np = (n + 1) & (NSLOT - 1);
            const float c = s0 * trow[nm] + s1 * trow[n] + s2 * trow[np];
            wo[n] = __powf(c, gm) * invS;
        }
    }
}

extern "C" void kernel_launch(void* const* d_in, const int* in_sizes, int n_in,
                              void* d_out, int out_size, void* d_ws, size_t ws_size,
                              hipStream_t stream) {
    (void)in_sizes; (void)n_in; (void)out_size; (void)d_ws; (void)ws_size;
    const float* memory = (const float*)d_in[0];
    const float* k      = (const float*)d_in[1];
    const float* beta   = (const float*)d_in[2];
    const float* g      = (const float*)d_in[3];
    const float* s      = (const float*)d_in[4];
    const float* gamma  = (const float*)d_in[5];
    const float* w_prev = (const float*)d_in[6];
    float* out = (float*)d_out;

    ntm_address_kernel<<<dim3(BATCH), dim3(256), 0, stream>>>(
        memory, k, beta, g, s, gamma, w_prev, out);
}